// ConcatHeadModule_88201448391414
// MI455X (gfx1250) — compile-verified
//
#include <hip/hip_runtime.h>
#include <hip/hip_bf16.h>

// ---------------------------------------------------------------------------
// ConcatHeadModule on gfx1250 (MI455X, wave32).
//   Phase 1: fp32 WMMA (V_WMMA_F32_16X16X4_F32) for the dense GEMMs.
//   Phase 2: hardware v_tanh_f32 + fmac over the 1024x1024x64 outer block.
// ---------------------------------------------------------------------------

typedef float v2f __attribute__((ext_vector_type(2)));
typedef float v8f __attribute__((ext_vector_type(8)));

#define NN    1024   // n
#define K1    512    // 2*LDIMS
#define HID   128
#define HID2  64

__device__ __forceinline__ float fast_tanh(float x) {
#if __has_builtin(__builtin_amdgcn_tanhf)
    return __builtin_amdgcn_tanhf(x);   // gfx1250 v_tanh_f32
#else
    return tanhf(x);
#endif
}

// D = A(16x4) x B(4x16) + C, fp32 WMMA.
// A frag (lane<16): {A[m=lane][k0], A[m=lane][k0+1]}; (lane>=16): k0+2,k0+3.
// B frag (lane<16): {B[k0][n=lane], B[k0+1][n=lane]}; (lane>=16): k0+2,k0+3.
// C/D: vgpr r holds out[m = r + 8*(lane>=16)][n = lane&15].
__device__ __forceinline__ v8f wmma_f32(v2f a, v2f b, v8f c) {
    return __builtin_amdgcn_wmma_f32_16x16x4_f32(
        /*neg_a=*/false, a, /*neg_b=*/false, b,
        /*c_mod=*/(short)0, c, /*reuse_a=*/false, /*reuse_b=*/false);
}

// ---------------------------------------------------------------------------
// Kernel A: hf = tanh(xf @ W_foh + cat_bias[:HID])   [1024 x 128]
//           hm = tanh(xf @ W_fom + cat_bias[HID:])   [1024 x 128]
// One wave per 16x16 output tile; K=512 -> 128 WMMA steps.
// tile id: [mat(2)][mtile(64)][ntile(8)]  -> 1024 waves.
// ---------------------------------------------------------------------------
__global__ __launch_bounds__(128) void gemm1_tanh_kernel(
    const float* __restrict__ xf,
    const float* __restrict__ W_foh, const float* __restrict__ W_fom,
    const float* __restrict__ cat_bias,
    float* __restrict__ hf, float* __restrict__ hm)
{
    const int tile = blockIdx.x * blockDim.y + threadIdx.y;  // 0..1023
    const int lane = threadIdx.x;                            // 0..31
    const int mat   = tile >> 9;        // 0 = foh, 1 = fom
    const int rem   = tile & 511;
    const int mtile = rem >> 3;         // 0..63
    const int ntile = rem & 7;          // 0..7

    const float* __restrict__ W    = mat ? W_fom : W_foh;
    float*       __restrict__ outp = mat ? hm    : hf;
    const float* __restrict__ bias = cat_bias + mat * HID;

    const int half = lane >> 4;         // 0|1
    const int l16  = lane & 15;
    const int row  = mtile * 16 + l16;
    const int col  = ntile * 16 + l16;
    const int khi  = half * 2;          // 0|2

    const float* arow = xf + row * K1 + khi;        // 8B aligned
    const float* bcol = W  + khi * HID + col;

    v8f c = {};
    #pragma unroll 4
    for (int k = 0; k < K1; k += 4) {
        v2f a = *reinterpret_cast<const v2f*>(arow + k);
        v2f b;
        b.x = bcol[k * HID];
        b.y = bcol[k * HID + HID];
        c = wmma_f32(a, b, c);
    }

    const float bn = bias[col];
    #pragma unroll
    for (int r = 0; r < 8; ++r) {
        const int m = mtile * 16 + r + half * 8;
        outp[m * HID + col] = fast_tanh(c[r] + bn);
    }
}

// ---------------------------------------------------------------------------
// Kernel B: th  = hf @ W_hid2[:128]  + hid2_bias    [1024 x 64], row-major
//           tmT = (hm @ W_hid2[128:])^T             [64 x 1024], transposed
// One wave per 16x16 tile; K=128 -> 32 WMMA steps. 512 waves total.
// ---------------------------------------------------------------------------
__global__ __launch_bounds__(128) void gemm2_kernel(
    const float* __restrict__ hf, const float* __restrict__ hm,
    const float* __restrict__ W_hid2, const float* __restrict__ hid2_bias,
    float* __restrict__ th, float* __restrict__ tmT)
{
    const int tile = blockIdx.x * blockDim.y + threadIdx.y;  // 0..511
    const int lane = threadIdx.x;
    const int mat   = tile >> 8;        // 0 = head(th), 1 = mod(tmT)
    const int rem   = tile & 255;
    const int mtile = rem >> 2;         // 0..63
    const int ntile = rem & 3;          // 0..3

    const float* __restrict__ A = mat ? hm : hf;
    const float* __restrict__ B = W_hid2 + mat * HID * HID2; // top/bottom block

    const int half = lane >> 4;
    const int l16  = lane & 15;
    const int row  = mtile * 16 + l16;
    const int col  = ntile * 16 + l16;
    const int khi  = half * 2;

    const float* arow = A + row * HID + khi;
    const float* bcol = B + khi * HID2 + col;

    v8f c = {};
    #pragma unroll
    for (int k = 0; k < HID; k += 4) {
        v2f a = *reinterpret_cast<const v2f*>(arow + k);
        v2f b;
        b.x = bcol[k * HID2];
        b.y = bcol[k * HID2 + HID2];
        c = wmma_f32(a, b, c);
    }

    if (mat == 0) {
        const float bn = hid2_bias[col];
        #pragma unroll
        for (int r = 0; r < 8; ++r) {
            const int m = mtile * 16 + r + half * 8;
            th[m * HID2 + col] = c[r] + bn;
        }
    } else {
        #pragma unroll
        for (int r = 0; r < 8; ++r) {
            const int m = mtile * 16 + r + half * 8;
            tmT[col * NN + m] = c[r];            // transposed store (once)
        }
    }
}

// ---------------------------------------------------------------------------
// Kernel C: scores[i][j] = sum_d tanh(th[i][d] + tmT[d][j]) * w[d] + out_bias
// 256 threads/block, 8 i-rows per thread. Per d: 1 coalesced tmT load,
// broadcast th/w from LDS, 8 x (v_tanh_f32 + v_fmac_f32).
// ---------------------------------------------------------------------------
#define IPT 8
__global__ __launch_bounds__(256) void outer_tanh_dot_kernel(
    const float* __restrict__ th, const float* __restrict__ tmT,
    const float* __restrict__ W_out, const float* __restrict__ out_bias,
    float* __restrict__ out)
{
    __shared__ float sw[HID2];
    __shared__ float sth[IPT][HID2];

    const int tx = threadIdx.x;
    const int j  = blockIdx.x * 256 + tx;
    const int i0 = blockIdx.y * IPT;

    if (tx < HID2) sw[tx] = W_out[tx];
    for (int t = tx; t < IPT * HID2; t += 256)
        sth[t >> 6][t & 63] = th[(i0 + (t >> 6)) * HID2 + (t & 63)];
    __syncthreads();

    float acc[IPT];
    #pragma unroll
    for (int ii = 0; ii < IPT; ++ii) acc[ii] = 0.0f;

    #pragma unroll 4
    for (int d = 0; d < HID2; ++d) {
        const float tmv = tmT[d * NN + j];   // unit-stride across lanes
        const float wd  = sw[d];             // LDS broadcast
        #pragma unroll
        for (int ii = 0; ii < IPT; ++ii)
            acc[ii] = fmaf(wd, fast_tanh(sth[ii][d] + tmv), acc[ii]);
    }

    const float ob = out_bias[0];
    #pragma unroll
    for (int ii = 0; ii < IPT; ++ii)
        out[(i0 + ii) * NN + j] = acc[ii] + ob;
}

// ---------------------------------------------------------------------------
extern "C" void kernel_launch(void* const* d_in, const int* in_sizes, int n_in,
                              void* d_out, int out_size, void* d_ws, size_t ws_size,
                              hipStream_t stream) {
    const float* xf        = (const float*)d_in[0]; // (1024,2,256) == (1024,512)
    const float* W_foh     = (const float*)d_in[1]; // (512,128)
    const float* W_fom     = (const float*)d_in[2]; // (512,128)
    const float* cat_bias  = (const float*)d_in[3]; // (256,)
    const float* W_hid2    = (const float*)d_in[4]; // (256,64)
    const float* hid2_bias = (const float*)d_in[5]; // (64,)
    const float* W_out     = (const float*)d_in[6]; // (64,)
    const float* out_bias  = (const float*)d_in[7]; // (1,)
    float* out = (float*)d_out;                     // (1024,1024)

    // Workspace layout (1.5 MB total, fully overwritten each call):
    float* hf  = (float*)d_ws;        // 1024*128
    float* hm  = hf  + NN * HID;      // 1024*128
    float* th  = hm  + NN * HID;      // 1024*64
    float* tmT = th  + NN * HID2;     // 64*1024

    // Phase 1a: 1024 tiles, 4 waves/block.
    gemm1_tanh_kernel<<<dim3(256), dim3(32, 4), 0, stream>>>(
        xf, W_foh, W_fom, cat_bias, hf, hm);

    // Phase 1b: 512 tiles, 4 waves/block.
    gemm2_kernel<<<dim3(128), dim3(32, 4), 0, stream>>>(
        hf, hm, W_hid2, hid2_bias, th, tmT);

    // Phase 2: 1024x1024 outputs, 8 rows per thread.
    outer_tanh_dot_kernel<<<dim3(NN / 256, NN / IPT), dim3(256), 0, stream>>>(
        th, tmT, W_out, out_bias, out);
}